// GCN_16673063043610
// MI455X (gfx1250) — compile-verified
//
#include <hip/hip_runtime.h>

#define Nn   50000
#define Ee   1600000
#define KDIM 128
#define EPSF 1e-5f

typedef __attribute__((ext_vector_type(2))) float v2f;
typedef __attribute__((ext_vector_type(8))) float v8f;

// ---------------- degree / normalization ----------------
__global__ __launch_bounds__(256) void deg_init(float* deg) {
    int i = blockIdx.x * 256 + threadIdx.x;
    if (i < Nn) deg[i] = 1.0f;                       // self loop
}

__global__ __launch_bounds__(256) void deg_count(const long long* __restrict__ ei,
                                                 float* __restrict__ deg) {
    int e = blockIdx.x * 256 + threadIdx.x;          // grid = Ee/256 exact
    int d = (int)ei[(size_t)Ee + e];
    atomicAdd(&deg[d], 1.0f);
}

__global__ __launch_bounds__(256) void deg_rsqrt(float* deg) {
    int i = blockIdx.x * 256 + threadIdx.x;
    if (i < Nn) deg[i] = rsqrtf(deg[i]);             // deg >= 1 always
}

// ---------------- WMMA GEMM: h = X @ W ; agg = h * dinv^2 (+bias) ----------------
// X: [Nn x 128] row-major, W: [128 x COUT] row-major.
// One wave computes a 16x16 tile with V_WMMA_F32_16X16X4_F32, K=128 -> 32 WMMAs.
template<int COUT, bool BIAS_IN_AGG>
__global__ __launch_bounds__((COUT / 16) * 32) void gemm_wmma(
        const float* __restrict__ X, const float* __restrict__ W,
        const float* __restrict__ bias, const float* __restrict__ dinv,
        float* __restrict__ h, float* __restrict__ agg) {
    const int wave = threadIdx.x >> 5;
    const int lane = threadIdx.x & 31;
    const int half = lane >> 4;      // 0: K pair {0,1}, 1: K pair {2,3}
    const int lm   = lane & 15;
    const int m0   = blockIdx.x * 16;    // grid = Nn/16 = 3125 exact
    const int n0   = wave * 16;

    v8f c = {};
    const float* arow = X + (size_t)(m0 + lm) * KDIM + 2 * half;
    const float* bcol = W + (size_t)(2 * half) * COUT + (n0 + lm);

#pragma unroll 4
    for (int k0 = 0; k0 < KDIM; k0 += 4) {
        v2f a = *(const v2f*)arow;               // A[lm][k0+2h], A[lm][k0+2h+1]
        v2f b;
        b.x = bcol[0];                           // B[k0+2h  ][n0+lm]
        b.y = bcol[COUT];                        // B[k0+2h+1][n0+lm]
        c = __builtin_amdgcn_wmma_f32_16x16x4_f32(
                /*neg_a=*/false, a, /*neg_b=*/false, b,
                /*c_mod=*/(short)0, c, /*reuse_a=*/false, /*reuse_b=*/false);
        arow += 4;
        bcol += 4 * COUT;
    }

#pragma unroll
    for (int v = 0; v < 8; ++v) {
        int m = m0 + v + 8 * half;               // D layout: lanes 16-31 -> rows 8-15
        int n = n0 + lm;
        float val = c[v];
        h[(size_t)m * COUT + n] = val;
        float sn = dinv[m];
        float aval = val * (sn * sn);            // self-loop contribution
        if (BIAS_IN_AGG) aval += bias[n];
        agg[(size_t)m * COUT + n] = aval;
    }
}

// ---------------- edge scatter: agg[dst] += h[src] * dinv[src]*dinv[dst] ----------------
__global__ __launch_bounds__(256) void scatter128(const long long* __restrict__ ei,
                                                  const float* __restrict__ dinv,
                                                  const float* __restrict__ h,
                                                  float* __restrict__ agg) {
    const int wave = threadIdx.x >> 5;
    const int lane = threadIdx.x & 31;
    const size_t e = (size_t)blockIdx.x * 8 + wave;   // grid = Ee/8 = 200000 exact
    const int s = (int)ei[e];
    const int d = (int)ei[(size_t)Ee + e];
    const float w = dinv[s] * dinv[d];
    const float4 v = *(const float4*)&h[(size_t)s * 128 + lane * 4];
    float* base = &agg[(size_t)d * 128 + lane * 4];
    atomicAdd(base + 0, v.x * w);
    atomicAdd(base + 1, v.y * w);
    atomicAdd(base + 2, v.z * w);
    atomicAdd(base + 3, v.w * w);
}

__global__ __launch_bounds__(256) void scatter32(const long long* __restrict__ ei,
                                                 const float* __restrict__ dinv,
                                                 const float* __restrict__ h,
                                                 float* __restrict__ out) {
    const int wave = threadIdx.x >> 5;
    const int lane = threadIdx.x & 31;
    const size_t e = (size_t)blockIdx.x * 8 + wave;   // grid = Ee/8
    const int s = (int)ei[e];
    const int d = (int)ei[(size_t)Ee + e];
    const float w = dinv[s] * dinv[d];
    float v = h[(size_t)s * 32 + lane];
    atomicAdd(&out[(size_t)d * 32 + lane], v * w);
}

// ---------------- fused bias + ReLU + InstanceNorm (per node, in place) ----------------
__global__ __launch_bounds__(256) void post_inorm(const float* __restrict__ bias,
                                                  float* __restrict__ buf) {
    const int wave = threadIdx.x >> 5;
    const int lane = threadIdx.x & 31;
    const int node = blockIdx.x * 8 + wave;          // grid = Nn/8 = 6250 exact
    float4 v = *(float4*)&buf[(size_t)node * 128 + lane * 4];
    const float4 b = *(const float4*)&bias[lane * 4];
    v.x = fmaxf(v.x + b.x, 0.0f);
    v.y = fmaxf(v.y + b.y, 0.0f);
    v.z = fmaxf(v.z + b.z, 0.0f);
    v.w = fmaxf(v.w + b.w, 0.0f);
    float s  = v.x + v.y + v.z + v.w;
    float ss = v.x * v.x + v.y * v.y + v.z * v.z + v.w * v.w;
#pragma unroll
    for (int off = 16; off >= 1; off >>= 1) {        // wave32 tree reduction
        s  += __shfl_xor(s,  off, 32);
        ss += __shfl_xor(ss, off, 32);
    }
    const float mean = s * (1.0f / 128.0f);
    const float var  = ss * (1.0f / 128.0f) - mean * mean;   // biased var, matches ref
    const float r    = rsqrtf(var + EPSF);
    v.x = (v.x - mean) * r;
    v.y = (v.y - mean) * r;
    v.z = (v.z - mean) * r;
    v.w = (v.w - mean) * r;
    *(float4*)&buf[(size_t)node * 128 + lane * 4] = v;
}

// ---------------- host launcher ----------------
extern "C" void kernel_launch(void* const* d_in, const int* in_sizes, int n_in,
                              void* d_out, int out_size, void* d_ws, size_t ws_size,
                              hipStream_t stream) {
    const float*     x  = (const float*)d_in[0];
    const long long* ei = (const long long*)d_in[1];
    const float*     W1 = (const float*)d_in[2];
    const float*     b1 = (const float*)d_in[3];
    const float*     W2 = (const float*)d_in[4];
    const float*     b2 = (const float*)d_in[5];
    const float*     W3 = (const float*)d_in[6];
    const float*     b3 = (const float*)d_in[7];
    float* out = (float*)d_out;

    float* bufA = (float*)d_ws;                      // h buffer        [Nn x 128]
    float* bufB = bufA + (size_t)Nn * 128;           // agg/input       [Nn x 128]
    float* bufC = bufB + (size_t)Nn * 128;           // agg/input       [Nn x 128]
    float* dinv = bufC + (size_t)Nn * 128;           // deg -> dinv     [Nn]

    // D^{-1/2} with self loops
    deg_init <<<(Nn + 255) / 256, 256, 0, stream>>>(dinv);
    deg_count<<<Ee / 256,        256, 0, stream>>>(ei, dinv);
    deg_rsqrt<<<(Nn + 255) / 256, 256, 0, stream>>>(dinv);

    // Layer 1: x -> bufA (h), bufB (agg incl self loop) -> scatter -> inorm in place
    gemm_wmma<128, false><<<Nn / 16, 256, 0, stream>>>(x, W1, nullptr, dinv, bufA, bufB);
    scatter128<<<Ee / 8, 256, 0, stream>>>(ei, dinv, bufA, bufB);
    post_inorm<<<Nn / 8, 256, 0, stream>>>(b1, bufB);

    // Layer 2: bufB -> bufA (h), bufC (agg) -> scatter -> inorm in place
    gemm_wmma<128, false><<<Nn / 16, 256, 0, stream>>>(bufB, W2, nullptr, dinv, bufA, bufC);
    scatter128<<<Ee / 8, 256, 0, stream>>>(ei, dinv, bufA, bufC);
    post_inorm<<<Nn / 8, 256, 0, stream>>>(b2, bufC);

    // Layer 3: bufC -> bufA (h3, Nn x 32), d_out initialized with self loop + bias
    gemm_wmma<32, true><<<Nn / 16, 64, 0, stream>>>(bufC, W3, b3, dinv, bufA, out);
    scatter32<<<Ee / 8, 256, 0, stream>>>(ei, dinv, bufA, out);
}